// Gen_IMP_47390669144623
// MI455X (gfx1250) — compile-verified
//
#include <hip/hip_runtime.h>

#define N_V 100000
#define N_E 100000
#define NP  2000000
#define FD  64

typedef __attribute__((ext_vector_type(2))) float v2f;
typedef __attribute__((ext_vector_type(8))) float v8f;

__global__ void zero_f32(float* __restrict__ p, long long n) {
  long long i = (long long)blockIdx.x * blockDim.x + threadIdx.x;
  if (i < n) p[i] = 0.0f;
}

// One thread per (pair, feature). 64 consecutive threads share one pair ->
// wave-uniform indices, 128B-contiguous gather and 128B-contiguous L2 atomics.
__global__ void scatter_add64(const float* __restrict__ src,
                              const int*   __restrict__ gidx,
                              const int*   __restrict__ sidx,
                              float* __restrict__ agg,
                              float* __restrict__ cnt,
                              int npairs) {
  long long t = (long long)blockIdx.x * blockDim.x + threadIdx.x;
  long long total = (long long)npairs * FD;
  if (t >= total) return;
  int p = (int)(t >> 6);
  int f = (int)(t & 63);
  int g = gidx[p];
  int s = sidx[p];
  atomicAdd(&agg[(long long)s * FD + f], src[(long long)g * FD + f]);
  if (f == 0) atomicAdd(&cnt[s], 1.0f);
}

__global__ void inv_count(float* __restrict__ cnt, int n) {
  int i = blockIdx.x * blockDim.x + threadIdx.x;
  if (i < n) cnt[i] = 1.0f / fmaxf(cnt[i], 1.0f);
}

// out[r,0:64] = relu( concat(Xa[r], Xb[r]) @ W(128x64) + bias ),
// with the half selected by scale_b multiplied per-row by inv[r] (segment mean).
// One wave per 16-row tile; f32 WMMA 16x16x4.
// W is staged in LDS K-pair interleaved: Wp[(k>>1)*128 + n*2 + (k&1)], so a
// B fragment {W[kk][n], W[kk+1][n]} is one aligned ds_load_b64 straight into
// an even VGPR pair (no repacking movs, conflict-free banks).
__global__ __launch_bounds__(256)
void concat_gemm_relu(const float* __restrict__ Xa,
                      const float* __restrict__ Xb,
                      const float* __restrict__ inv,
                      int scale_b,
                      const float* __restrict__ W,     // [128,64] row-major
                      const float* __restrict__ bias,  // [64]
                      float* __restrict__ out,         // [rows,64]
                      int ntiles) {
  __shared__ float Wp[128 * 64];
  __shared__ float Bl[64];
  for (int i = threadIdx.x; i < 128 * 64; i += 256) {
    const int k = i >> 6;
    const int n = i & 63;
    Wp[(k >> 1) * 128 + n * 2 + (k & 1)] = W[i];
  }
  if (threadIdx.x < 64) Bl[threadIdx.x] = bias[threadIdx.x];
  __syncthreads();

  const int wave = threadIdx.x >> 5;
  const int tile = blockIdx.x * 8 + wave;
  if (tile >= ntiles) return;   // whole-wave predication: EXEC stays all-1s

  const int lane = threadIdx.x & 31;
  const int hw   = lane >> 4;     // half-wave select (K sub-pair)
  const int l15  = lane & 15;
  const int r0   = tile * 16;
  const int rowA = r0 + l15;

  const float sA = scale_b ? 1.0f : inv[rowA];
  const float sB = scale_b ? inv[rowA] : 1.0f;

  v8f acc[4];
#pragma unroll
  for (int t = 0; t < 4; ++t) {
    float bv = Bl[t * 16 + l15];
#pragma unroll
    for (int v = 0; v < 8; ++v) acc[t][v] = bv;
  }

  const float* Ra = Xa + (long long)rowA * FD;
  const float* Rb = Xb + (long long)rowA * FD;

  v2f af[16];

  // ---- Half 0: K = 0..63 from Xa ----
#pragma unroll
  for (int i = 0; i < 16; ++i) {
    const int kk = i * 4 + 2 * hw;
    af[i].x = Ra[kk] * sA;
    af[i].y = Ra[kk + 1] * sA;
  }
#pragma unroll
  for (int i = 0; i < 16; ++i) {
    const int kp = i * 2 + hw;          // K-pair index = kk>>1
#pragma unroll
    for (int t = 0; t < 4; ++t) {
      const int n = t * 16 + l15;
      const v2f b = *(const v2f*)&Wp[kp * 128 + n * 2];
      acc[t] = __builtin_amdgcn_wmma_f32_16x16x4_f32(
          false, af[i], false, b, (short)0, acc[t], false, false);
    }
  }

  // ---- Half 1: K = 64..127 from Xb ----
#pragma unroll
  for (int i = 0; i < 16; ++i) {
    const int kk = i * 4 + 2 * hw;
    af[i].x = Rb[kk] * sB;
    af[i].y = Rb[kk + 1] * sB;
  }
#pragma unroll
  for (int i = 0; i < 16; ++i) {
    const int kp = 32 + i * 2 + hw;
#pragma unroll
    for (int t = 0; t < 4; ++t) {
      const int n = t * 16 + l15;
      const v2f b = *(const v2f*)&Wp[kp * 128 + n * 2];
      acc[t] = __builtin_amdgcn_wmma_f32_16x16x4_f32(
          false, af[i], false, b, (short)0, acc[t], false, false);
    }
  }

#pragma unroll
  for (int t = 0; t < 4; ++t) {
    const int col = t * 16 + l15;
#pragma unroll
    for (int v = 0; v < 8; ++v) {
      const int row = r0 + v + 8 * hw;
      out[(long long)row * FD + col] = fmaxf(acc[t][v], 0.0f);
    }
  }
}

// Last E2V layer: out dim 1 -> plain dot, weights broadcast from LDS.
__global__ void last_linear(const float* __restrict__ aggv,
                            const float* __restrict__ inv,
                            const float* __restrict__ hn,
                            const float* __restrict__ w,  // [128]
                            const float* __restrict__ b,  // [1]
                            float* __restrict__ out, int rows) {
  __shared__ float wl[128];
  if (threadIdx.x < 128) wl[threadIdx.x] = w[threadIdx.x];
  __syncthreads();
  int r = blockIdx.x * blockDim.x + threadIdx.x;
  if (r >= rows) return;
  float s = b[0];
  const float iv = inv[r];
  const float* ra = aggv + (long long)r * FD;
  const float* rh = hn   + (long long)r * FD;
#pragma unroll 8
  for (int k = 0; k < 64; ++k) s += ra[k] * iv * wl[k];
#pragma unroll 8
  for (int k = 0; k < 64; ++k) s += rh[k] * wl[64 + k];
  out[r] = fmaxf(s, 0.0f);
}

extern "C" void kernel_launch(void* const* d_in, const int* in_sizes, int n_in,
                              void* d_out, int out_size, void* d_ws, size_t ws_size,
                              hipStream_t stream) {
  const float* he_in = (const float*)d_in[0];
  const float* hn_in = (const float*)d_in[1];
  const int*   aff   = (const int*)d_in[2];
  const int* node_idx = aff;        // ve_affiliation[0]
  const int* edge_idx = aff + NP;   // ve_affiliation[1]
  const float* v2e_w[3] = {(const float*)d_in[3], (const float*)d_in[7],  (const float*)d_in[11]};
  const float* v2e_b[3] = {(const float*)d_in[4], (const float*)d_in[8],  (const float*)d_in[12]};
  const float* e2v_w[3] = {(const float*)d_in[5], (const float*)d_in[9],  (const float*)d_in[13]};
  const float* e2v_b[3] = {(const float*)d_in[6], (const float*)d_in[10], (const float*)d_in[14]};
  float* out = (float*)d_out;

  // Workspace layout: he | hn | agg (shared e/v) | cnt (shared e/v)
  float* he  = (float*)d_ws;
  float* hn  = he  + (size_t)N_E * FD;
  float* agg = hn  + (size_t)N_V * FD;
  float* cnt = agg + (size_t)N_E * FD;

  hipMemcpyAsync(he, he_in, sizeof(float) * (size_t)N_E * FD, hipMemcpyDeviceToDevice, stream);
  hipMemcpyAsync(hn, hn_in, sizeof(float) * (size_t)N_V * FD, hipMemcpyDeviceToDevice, stream);

  const long long featTot = (long long)NP * FD;
  const int sBlocks  = (int)((featTot + 255) / 256);
  const int ntE = N_E / 16, ntV = N_V / 16;      // both divide exactly
  const int gBlocksE = (ntE + 7) / 8, gBlocksV = (ntV + 7) / 8;

  for (int l = 0; l < 3; ++l) {
    // ---- V2E: nodes -> edges ----
    zero_f32<<<(N_E * FD + 255) / 256, 256, 0, stream>>>(agg, (long long)N_E * FD);
    zero_f32<<<(N_E + 255) / 256, 256, 0, stream>>>(cnt, N_E);
    scatter_add64<<<sBlocks, 256, 0, stream>>>(hn, node_idx, edge_idx, agg, cnt, NP);
    inv_count<<<(N_E + 255) / 256, 256, 0, stream>>>(cnt, N_E);
    concat_gemm_relu<<<gBlocksE, 256, 0, stream>>>(he, agg, cnt, /*scale_b=*/1,
                                                   v2e_w[l], v2e_b[l], he, ntE);
    // ---- E2V: edges -> nodes ----
    zero_f32<<<(N_V * FD + 255) / 256, 256, 0, stream>>>(agg, (long long)N_V * FD);
    zero_f32<<<(N_V + 255) / 256, 256, 0, stream>>>(cnt, N_V);
    scatter_add64<<<sBlocks, 256, 0, stream>>>(he, edge_idx, node_idx, agg, cnt, NP);
    inv_count<<<(N_V + 255) / 256, 256, 0, stream>>>(cnt, N_V);
    if (l < 2) {
      concat_gemm_relu<<<gBlocksV, 256, 0, stream>>>(agg, hn, cnt, /*scale_b=*/0,
                                                     e2v_w[l], e2v_b[l], hn, ntV);
    } else {
      last_linear<<<(N_V + 255) / 256, 256, 0, stream>>>(agg, cnt, hn, e2v_w[2], e2v_b[2],
                                                         out + (size_t)N_E * FD, N_V);
    }
  }
  // Final hyperedge features -> first N_E*64 floats of d_out
  hipMemcpyAsync(out, he, sizeof(float) * (size_t)N_E * FD, hipMemcpyDeviceToDevice, stream);
}